// IO_Layer_1700807049813
// MI455X (gfx1250) — compile-verified
//
#include <hip/hip_runtime.h>

// Problem sizes (fixed by the reference)
#define NB   32      // batch
#define NIN  1024    // input width
#define NM   4096    // memory rows
#define NE   1024    // memory width
#define NOUT 1024    // output width

typedef __attribute__((ext_vector_type(2))) float v2f;
typedef __attribute__((ext_vector_type(8))) float v8f;
typedef unsigned int u32;
typedef __attribute__((ext_vector_type(4))) u32 v4u;
typedef __attribute__((ext_vector_type(8))) int  v8i;
typedef __attribute__((ext_vector_type(4))) int  v4i;

// ---------------------------------------------------------------------------
// WMMA f32 16x16x4 fragment layouts (CDNA5 ISA 7.12.2, wave32):
//   A (16x4, MxK):  lane l -> m = l&15, kbase = 2*(l>>4); regs {A[m][kb], A[m][kb+1]}
//   B (4x16, KxN):  lane l -> n = l&15, kbase = 2*(l>>4); regs {B[kb][n], B[kb+1][n]}
//   C/D (16x16):    lane l -> n = l&15, reg i -> row = i + 8*(l>>4)
// Each 128-thread block = 4 wave32s; each wave owns one 16x16 output tile.
// ---------------------------------------------------------------------------

// TDM: issue a 2-D tensor_load_to_lds (6-arg builtin on this toolchain:
//   (uint32x4 g0, int32x8 g1, int32x4 g2, int32x4 g3, int32x8 extra, i32 cpol)).
//   Loads tile_rows x tile_cols elements of 4-byte data, row stride = row_stride
//   elements in global memory, optionally inserting 1 DWORD of LDS padding per
//   64 DWORDs (so a 64-wide f32 row lands at stride 65 -> bank-conflict-free).
// D# layout per CDNA5 ISA 8.3/8.4 (group0 128b, group1 256b; groups 2/3 unused
// for 2-D tiles and passed as zeros).
__device__ __forceinline__ void tdm_load_2d_f32(const float* gptr, u32 lds_byte_off,
                                                int tile_cols, int tile_rows,
                                                long row_stride, bool pad64) {
  unsigned long long ga = (unsigned long long)gptr;
  v4u g0;
  g0.x = 1u;                                   // count=1, user descriptor
  g0.y = lds_byte_off;                         // bits 63:32  lds_addr (bytes)
  g0.z = (u32)(ga & 0xffffffffu);              // bits 95:64  global_addr[31:0]
  g0.w = (u32)((ga >> 32) & 0x01ffffffu)       // bits 120:96 global_addr[56:32]
       | (2u << 30);                           // bits 127:126 type=2 (image)
  u32 w0 = (2u << 16);                         // data_size = 2 -> 4 bytes
  if (pad64) w0 |= (1u << 20)                  // pad_enable
                 | (5u << 22)                  // pad_interval: 64 DWORDs
                 | (0u << 25);                 // pad_amount: 1 DWORD
  u32 d0 = (u32)tile_cols;                     // tensor_dim0 == tile width (no OOB)
  u32 d1 = (u32)tile_rows;                     // tensor_dim1 == tile height
  unsigned long long st = (unsigned long long)row_stride;  // dim0 stride (elems)
  v8i g1;
  g1[0] = (int)w0;                                            // mask=0, flags
  g1[1] = (int)((d0 & 0xffffu) << 16);                        // [31:16]=dim0 lo16
  g1[2] = (int)((d0 >> 16) | ((d1 & 0xffffu) << 16));         // dim0 hi16 | dim1 lo16
  g1[3] = (int)(((d1 >> 16) & 0xffffu) | ((u32)tile_cols << 16)); // dim1 hi | tile_dim0
  g1[4] = (int)((u32)tile_rows & 0xffffu);                    // tile_dim1 | tile_dim2=0
  g1[5] = (int)(u32)(st & 0xffffffffu);                       // dim0_stride[31:0]
  g1[6] = (int)(u32)((st >> 32) & 0xffffu);                   // dim0_stride[47:32]
  g1[7] = 0;                                                  // dim1_stride (unused, 2-D)
  v4i z4 = {0, 0, 0, 0};
  v8i z8 = {0, 0, 0, 0, 0, 0, 0, 0};
  __builtin_amdgcn_tensor_load_to_lds(g0, g1, z4, z4, z8, 0);
}

// Kernel 1: proj[z][b][e] = x(32x1024) @ W_z(1024x1024), z in {w_kg, r_kg, w_evg, w_vg}
__global__ void proj_gemm(const float* __restrict__ x,
                          const float* __restrict__ w_kg,
                          const float* __restrict__ r_kg,
                          const float* __restrict__ w_evg,
                          const float* __restrict__ w_vg,
                          float* __restrict__ proj) {
  const int lane = threadIdx.x & 31;
  const int wave = threadIdx.x >> 5;
  const int n  = (blockIdx.x * 4 + wave) * 16 + (lane & 15);
  const int m  = blockIdx.y * 16 + (lane & 15);
  const int kh = (lane >> 4) * 2;
  const float* W = (blockIdx.z == 0) ? w_kg : (blockIdx.z == 1) ? r_kg
                 : (blockIdx.z == 2) ? w_evg : w_vg;
  v8f acc = {};
  for (int k0 = 0; k0 < NIN; k0 += 4) {
    const int ka = k0 + kh;
    v2f a, b;
    a.x = x[m * NIN + ka];
    a.y = x[m * NIN + ka + 1];
    b.x = W[(size_t)(ka + 0) * NE + n];
    b.y = W[(size_t)(ka + 1) * NE + n];
    if ((k0 & 63) == 0 && (k0 + 64) < NIN)
      __builtin_prefetch(&W[(size_t)(ka + 64) * NE + n], 0, 0);
    acc = __builtin_amdgcn_wmma_f32_16x16x4_f32(false, a, false, b, (short)0,
                                                acc, false, false);
  }
  float* o = proj + (size_t)blockIdx.z * NB * NE;
  const int rbase = blockIdx.y * 16 + (lane >> 4) * 8;
#pragma unroll
  for (int i = 0; i < 8; ++i) o[(size_t)(rbase + i) * NE + n] = acc[i];
}

// Kernel 2: rninv[m] = 1 / ||memory[m,:]||
__global__ void rownorm_inv(const float* __restrict__ mem, float* __restrict__ rninv) {
  __shared__ float red[256];
  const int row = blockIdx.x, t = threadIdx.x;
  float s = 0.f;
  for (int e = t; e < NE; e += 256) {
    float v = mem[(size_t)row * NE + e];
    s += v * v;
  }
  red[t] = s; __syncthreads();
  for (int o = 128; o > 0; o >>= 1) {
    if (t < o) red[t] += red[t + o];
    __syncthreads();
  }
  if (t == 0) rninv[row] = rsqrtf(red[0]);
}

// Kernel 3: sims[z][b][m] = (key_z[b,:] . memory[m,:]) * rninv[m], z in {w,r}
__global__ void sims_gemm(const float* __restrict__ proj, const float* __restrict__ mem,
                          const float* __restrict__ rninv, float* __restrict__ sims) {
  const int lane = threadIdx.x & 31;
  const int wave = threadIdx.x >> 5;
  const int n  = (blockIdx.x * 4 + wave) * 16 + (lane & 15);  // memory row index
  const int m  = blockIdx.y * 16 + (lane & 15);               // batch index
  const int kh = (lane >> 4) * 2;
  const float* A = proj + (size_t)blockIdx.z * NB * NE;       // z=0: w_key, z=1: r_key
  v8f acc = {};
  for (int k0 = 0; k0 < NE; k0 += 4) {
    const int ka = k0 + kh;
    v2f a, b;
    a.x = A[m * NE + ka];
    a.y = A[m * NE + ka + 1];
    b.x = mem[(size_t)n * NE + ka];        // B[k][n] = memory[n][k] (implicit transpose)
    b.y = mem[(size_t)n * NE + ka + 1];
    acc = __builtin_amdgcn_wmma_f32_16x16x4_f32(false, a, false, b, (short)0,
                                                acc, false, false);
  }
  const float rs = rninv[n];
  float* o = sims + (size_t)blockIdx.z * NB * NM;
  const int rbase = blockIdx.y * 16 + (lane >> 4) * 8;
#pragma unroll
  for (int i = 0; i < 8; ++i) o[(size_t)(rbase + i) * NM + n] = acc[i] * rs;
}

// Kernel 4: per-batch normalize over M; emit r_weight, rw = r*w, s[b] = sum_m rw
__global__ void norm_weights(const float* __restrict__ sims, float* __restrict__ rn,
                             float* __restrict__ rw, float* __restrict__ svec) {
  __shared__ float r0[256], r1[256];
  __shared__ float inv0, inv1;
  const int b = blockIdx.x, t = threadIdx.x;
  const float* sw = sims + (size_t)b * NM;
  const float* sr = sims + (size_t)NB * NM + (size_t)b * NM;
  float s0 = 0.f, s1 = 0.f;
  for (int m = t; m < NM; m += 256) { s0 += sw[m]; s1 += sr[m]; }
  r0[t] = s0; r1[t] = s1; __syncthreads();
  for (int o = 128; o > 0; o >>= 1) {
    if (t < o) { r0[t] += r0[t + o]; r1[t] += r1[t + o]; }
    __syncthreads();
  }
  if (t == 0) { inv0 = 1.f / r0[0]; inv1 = 1.f / r1[0]; }
  __syncthreads();
  const float i0 = inv0, i1 = inv1;
  float sacc = 0.f;
  for (int m = t; m < NM; m += 256) {
    float w = sw[m] * i0;
    float r = sr[m] * i1;
    rn[(size_t)b * NM + m] = r;
    float p = w * r;
    rw[(size_t)b * NM + m] = p;
    sacc += p;
  }
  __syncthreads();
  r0[t] = sacc; __syncthreads();
  for (int o = 128; o > 0; o >>= 1) {
    if (t < o) r0[t] += r0[t + o];
    __syncthreads();
  }
  if (t == 0) svec[b] = r0[0];
}

// Kernel 5 (TDM pipeline): a2c2[z][b][e] = weights_z(32x4096) @ memory(4096x1024)
//   z=0: rn (r_weight), z=1: rw (r_weight*w_weight).
// memory is streamed as 64(K) x 64(N) f32 slabs via the Tensor Data Mover into
// double-buffered LDS (rows padded to stride 65 via TDM pad), overlapped with
// the 16 WMMAs per tile. Wave 0 issues the DMA; TENSORcnt + barrier fence it.
#define KT 64
#define LROW 65     // 64 + 1 DWORD TDM pad
__global__ void read_gemm_tdm(const float* __restrict__ rn, const float* __restrict__ rw,
                              const float* __restrict__ mem, float* __restrict__ a2c2) {
  __shared__ float smem[2][KT * LROW];
  const int lane = threadIdx.x & 31;
  const int wave = threadIdx.x >> 5;
  const int ncol0 = blockIdx.x * 64;                   // 64-wide slab of e
  const int nrel  = wave * 16 + (lane & 15);           // column within slab
  const int n     = ncol0 + nrel;
  const int m     = blockIdx.y * 16 + (lane & 15);     // b
  const int kh    = (lane >> 4) * 2;
  const float* A  = (blockIdx.z == 0) ? rn : rw;       // (B, M), K = NM
  const u32 lds0 = (u32)(size_t)&smem[0][0];
  const u32 lds1 = (u32)(size_t)&smem[1][0];

  if (wave == 0)   // prologue: DMA tile 0 into buffer 0
    tdm_load_2d_f32(mem + ncol0, lds0, 64, KT, NE, true);

  v8f acc = {};
  for (int t = 0; t < NM / KT; ++t) {
    const int buf = t & 1;
    if (wave == 0) {
      __builtin_amdgcn_s_wait_tensorcnt(0);            // tile t resident in LDS
      if (t + 1 < NM / KT)                             // prefetch tile t+1 via TDM
        tdm_load_2d_f32(mem + (size_t)(t + 1) * KT * NE + ncol0,
                        buf ? lds0 : lds1, 64, KT, NE, true);
    }
    __syncthreads();                                   // tile t visible to all waves
    const float* Bs = &smem[buf][0];
#pragma unroll 4
    for (int kk = 0; kk < KT; kk += 4) {
      const int kr = kk + kh;
      const int kg = t * KT + kr;
      v2f a, b;
      a.x = A[(size_t)m * NM + kg];
      a.y = A[(size_t)m * NM + kg + 1];
      b.x = Bs[(kr + 0) * LROW + nrel];
      b.y = Bs[(kr + 1) * LROW + nrel];
      acc = __builtin_amdgcn_wmma_f32_16x16x4_f32(false, a, false, b, (short)0,
                                                  acc, false, false);
    }
    __syncthreads();                                   // done reading before overwrite
  }
  float* o = a2c2 + (size_t)blockIdx.z * NB * NE;
  const int rbase = blockIdx.y * 16 + (lane >> 4) * 8;
#pragma unroll
  for (int i = 0; i < 8; ++i) o[(size_t)(rbase + i) * NE + n] = acc[i];
}

// Kernel 6: r_vect = A2 - erase*C2 + vect*s[b]
__global__ void rvect_combine(const float* __restrict__ a2c2, const float* __restrict__ proj,
                              const float* __restrict__ svec, float* __restrict__ rvect) {
  const int i = blockIdx.x * blockDim.x + threadIdx.x;  // < NB*NE
  const int b = i / NE;
  const float A2 = a2c2[i];
  const float C2 = a2c2[(size_t)NB * NE + i];
  const float er = proj[(size_t)2 * NB * NE + i];  // erase = x @ w_evg
  const float ve = proj[(size_t)3 * NB * NE + i];  // vect  = x @ w_vg
  rvect[i] = A2 - er * C2 + ve * svec[b];
}

// Kernel 7: out = [x | r_vect] (32x2048) @ post (2048x1024)
__global__ void final_gemm(const float* __restrict__ x, const float* __restrict__ rvect,
                           const float* __restrict__ post, float* __restrict__ out) {
  const int lane = threadIdx.x & 31;
  const int wave = threadIdx.x >> 5;
  const int n  = (blockIdx.x * 4 + wave) * 16 + (lane & 15);
  const int m  = blockIdx.y * 16 + (lane & 15);
  const int kh = (lane >> 4) * 2;
  v8f acc = {};
  for (int k0 = 0; k0 < NIN + NE; k0 += 4) {
    const int ka = k0 + kh;
    v2f a, b;
    a.x = (ka     < NIN) ? x[m * NIN + ka]     : rvect[m * NE + (ka - NIN)];
    a.y = (ka + 1 < NIN) ? x[m * NIN + ka + 1] : rvect[m * NE + (ka + 1 - NIN)];
    b.x = post[(size_t)(ka + 0) * NOUT + n];
    b.y = post[(size_t)(ka + 1) * NOUT + n];
    if ((k0 & 63) == 0 && (k0 + 64) < NIN + NE)
      __builtin_prefetch(&post[(size_t)(ka + 64) * NOUT + n], 0, 0);
    acc = __builtin_amdgcn_wmma_f32_16x16x4_f32(false, a, false, b, (short)0,
                                                acc, false, false);
  }
  const int rbase = blockIdx.y * 16 + (lane >> 4) * 8;
#pragma unroll
  for (int i = 0; i < 8; ++i) out[(size_t)(rbase + i) * NOUT + n] = acc[i];
}

extern "C" void kernel_launch(void* const* d_in, const int* in_sizes, int n_in,
                              void* d_out, int out_size, void* d_ws, size_t ws_size,
                              hipStream_t stream) {
  (void)in_sizes; (void)n_in; (void)out_size; (void)ws_size;
  const float* x     = (const float*)d_in[0];
  const float* mem   = (const float*)d_in[1];
  const float* w_kg  = (const float*)d_in[2];
  const float* w_vg  = (const float*)d_in[3];
  const float* w_evg = (const float*)d_in[4];
  const float* r_kg  = (const float*)d_in[5];
  const float* post  = (const float*)d_in[6];
  float* out = (float*)d_out;

  // Workspace carve-up (~3.0 MB of fp32)
  float* ws    = (float*)d_ws;
  float* proj  = ws;                               // 4*NB*NE  (w_key, r_key, erase, vect)
  float* sims  = proj  + (size_t)4 * NB * NE;      // 2*NB*NM
  float* rninv = sims  + (size_t)2 * NB * NM;      // NM
  float* rn    = rninv + NM;                       // NB*NM   (r_weight)
  float* rw    = rn    + (size_t)NB * NM;          // NB*NM   (r_weight * w_weight)
  float* svec  = rw    + (size_t)NB * NM;          // NB
  float* a2c2  = svec  + NB;                       // 2*NB*NE
  float* rvect = a2c2  + (size_t)2 * NB * NE;      // NB*NE

  proj_gemm    <<<dim3(NE / 64, NB / 16, 4), 128, 0, stream>>>(x, w_kg, r_kg, w_evg, w_vg, proj);
  rownorm_inv  <<<NM, 256, 0, stream>>>(mem, rninv);
  sims_gemm    <<<dim3(NM / 64, NB / 16, 2), 128, 0, stream>>>(proj, mem, rninv, sims);
  norm_weights <<<NB, 256, 0, stream>>>(sims, rn, rw, svec);
  read_gemm_tdm<<<dim3(NE / 64, NB / 16, 2), 128, 0, stream>>>(rn, rw, mem, a2c2);
  rvect_combine<<<(NB * NE) / 256, 256, 0, stream>>>(a2c2, proj, svec, rvect);
  final_gemm   <<<dim3(NOUT / 64, NB / 16, 1), 128, 0, stream>>>(x, rvect, post, out);
}